// EGATBaseModule_55241869361491
// MI455X (gfx1250) — compile-verified
//
#include <hip/hip_runtime.h>
#include <math.h>
#include <stdint.h>

// Problem constants (match reference).
#define NN   20000
#define EEDG 320000
#define IN_F 256
#define EIN_F 64
#define HH   4
#define FOUT 64       // per-head out feats (H*FOUT == IN_F)
#define FEDG 16       // per-head edge feats (H*FEDG == 64)
#define NEG_SLOPE 0.2f

typedef __attribute__((ext_vector_type(2))) float v2f;
typedef __attribute__((ext_vector_type(8))) float v8f;

// ---------------------------------------------------------------------------
// FP32 WMMA GEMM: C[M,Nc] = A[M,K] @ B[K,Nc], processing a 64-column slice
// of B per block (blockIdx.y selects the slice).
//
//  * Block = 256 threads = 8 waves. The block async-stages B[K x 64] into LDS
//    (GLOBAL_LOAD_ASYNC_TO_LDS_B32, drained with s_wait_asynccnt) using a
//    pair-major swizzle: element (k,c) -> Bs[(((k>>1)*16 + (c&15))*4 + (c>>4))*2
//    + (k&1)], so each 16x16 tile's B fragment (rows k0+2*hi, k0+2*hi+1 at one
//    column) is a single contiguous ds_load_b64.
//  * Each wave owns a 16-row strip and reuses its A fragment across the 4
//    N-tiles of the slice: per K-step = 1 global b64 + 4 ds b64 + 4 WMMAs.
//  * Requires M%16==0, K%4==0 (true here); strip guard is wave-uniform so
//    EXEC is all-1s around every V_WMMA.
// ---------------------------------------------------------------------------
template <int K>
__global__ __launch_bounds__(256) void egat_wmma_gemm(
    const float* __restrict__ A, const float* __restrict__ B,
    float* __restrict__ C, int M, int Nc) {
  __shared__ float Bs[K * 64];                 // 64KB (K=256) / 16KB (K=64)

  const int lane = threadIdx.x & 31;
  const int wave = threadIdx.x >> 5;
  const int l15  = lane & 15;
  const int hi   = lane >> 4;
  const int n0   = blockIdx.y * 64;

  // ---- Stage B[0:K, n0:n0+64] into LDS via async scatter (swizzled) ----
  const unsigned lds_base = (unsigned)(uintptr_t)(&Bs[0]);
#pragma unroll 4
  for (int t = threadIdx.x; t < K * 64; t += 256) {
    const int k = t >> 6;
    const int c = t & 63;
    const unsigned lds_off =
        lds_base + 4u * ((((unsigned)(k >> 1) * 16 + (c & 15)) * 4 + (c >> 4)) * 2 + (k & 1));
    const float* gp = B + (size_t)k * Nc + n0 + c;
    asm volatile("global_load_async_to_lds_b32 %0, %1, off"
                 :
                 : "v"(lds_off), "v"(gp)
                 : "memory");
  }
  asm volatile("s_wait_asynccnt 0x0" ::: "memory");
  __syncthreads();

  // ---- Compute: one 16-row strip per wave, 4 accumulator tiles wide ----
  const int strips = M >> 4;
  const int strip  = blockIdx.x * 8 + wave;    // wave-uniform
  if (strip < strips) {
    const int m0 = strip << 4;
    // A-frag: lane l15 -> row m0+l15 ; vgpr j -> K = k0 + 2*hi + j
    const float* Arow = A + (size_t)(m0 + l15) * K + 2 * hi;
    const v2f* Bsv = (const v2f*)Bs;

    v8f acc[4] = {{}, {}, {}, {}};
    for (int k0 = 0; k0 < K; k0 += 4) {
      v2f a = *(const v2f*)(Arow + k0);
      const int p = (k0 >> 1) + hi;            // row-pair index
      const v2f* Bp = Bsv + ((p * 16 + l15) << 2);
      v2f b0 = Bp[0], b1 = Bp[1], b2 = Bp[2], b3 = Bp[3];
      acc[0] = __builtin_amdgcn_wmma_f32_16x16x4_f32(false, a, false, b0, (short)0, acc[0], false, false);
      acc[1] = __builtin_amdgcn_wmma_f32_16x16x4_f32(false, a, false, b1, (short)0, acc[1], false, false);
      acc[2] = __builtin_amdgcn_wmma_f32_16x16x4_f32(false, a, false, b2, (short)0, acc[2], false, false);
      acc[3] = __builtin_amdgcn_wmma_f32_16x16x4_f32(false, a, false, b3, (short)0, acc[3], false, false);
    }
    // C/D layout: vgpr r -> row m0 + r + 8*hi, col n0 + nt*16 + l15
    float* Cb = C + (size_t)(m0 + 8 * hi) * Nc + n0 + l15;
#pragma unroll
    for (int nt = 0; nt < 4; ++nt)
#pragma unroll
      for (int r = 0; r < 8; ++r)
        Cb[(size_t)r * Nc + nt * 16] = acc[nt][r];
  }
}

// ---------------------------------------------------------------------------
// Init: out[:N*256] = x (residual base); m = -inf; denom = 0
// ---------------------------------------------------------------------------
__global__ __launch_bounds__(256) void egat_init(
    const float* __restrict__ x, float* __restrict__ out,
    float* __restrict__ m, float* __restrict__ denom, int nOut, int nNH) {
  int i = blockIdx.x * blockDim.x + threadIdx.x;
  if (i < nOut) out[i] = x[i];
  if (i < nNH) { m[i] = -INFINITY; denom[i] = 0.0f; }
}

// ---------------------------------------------------------------------------
// Edge scores: f = LeakyReLU(ni[src] + fij + nj[dst] + bias)   (fij preloaded
// into the f output region by the GEMM); e[edge,h] = dot(f_h, attn_h).
// ---------------------------------------------------------------------------
__device__ __forceinline__ float lrelu(float v) {
  return v >= 0.0f ? v : NEG_SLOPE * v;
}

__global__ __launch_bounds__(256) void egat_edge_scores(
    const int* __restrict__ src, const int* __restrict__ dst,
    const float4* __restrict__ ni, const float4* __restrict__ nj,
    const float4* __restrict__ attn, const float4* __restrict__ bias,
    float4* __restrict__ f, float* __restrict__ ebuf, int E) {
  int e = blockIdx.x * blockDim.x + threadIdx.x;
  if (e >= E) return;
  const int s = src[e], d = dst[e];
  const float4* nis = ni + (size_t)s * 16;
  const float4* njd = nj + (size_t)d * 16;
  float4* fe = f + (size_t)e * 16;
  float eh[HH] = {0.f, 0.f, 0.f, 0.f};
#pragma unroll
  for (int j = 0; j < 16; ++j) {
    float4 v = fe[j];
    float4 a = nis[j], b = njd[j], bi = bias[j];
    float4 r;
    r.x = lrelu(v.x + a.x + b.x + bi.x);
    r.y = lrelu(v.y + a.y + b.y + bi.y);
    r.z = lrelu(v.z + a.z + b.z + bi.z);
    r.w = lrelu(v.w + a.w + b.w + bi.w);
    fe[j] = r;
    float4 at = attn[j];
    eh[j >> 2] += r.x * at.x + r.y * at.y + r.z * at.z + r.w * at.w;
  }
#pragma unroll
  for (int h = 0; h < HH; ++h) ebuf[(size_t)e * HH + h] = eh[h];
}

// ---------------------------------------------------------------------------
// Segment max over dst (float max via signed/unsigned int atomic trick).
// ---------------------------------------------------------------------------
__device__ __forceinline__ void atomicMaxF32(float* addr, float val) {
  if (val >= 0.0f)
    atomicMax((int*)addr, __float_as_int(val));
  else
    atomicMin((unsigned int*)addr, __float_as_uint(val));
}

__global__ __launch_bounds__(256) void egat_segmax(
    const int* __restrict__ dst, const float* __restrict__ ebuf,
    float* __restrict__ m, int EH) {
  int i = blockIdx.x * blockDim.x + threadIdx.x;
  if (i >= EH) return;
  const int e = i >> 2, h = i & 3;
  atomicMaxF32(&m[(size_t)dst[e] * HH + h], ebuf[i]);
}

// ---------------------------------------------------------------------------
// ex = exp(e - m[dst]); denom[dst] += ex; ebuf overwritten with ex.
// ---------------------------------------------------------------------------
__global__ __launch_bounds__(256) void egat_expsum(
    const int* __restrict__ dst, float* __restrict__ ebuf,
    const float* __restrict__ m, float* __restrict__ denom, int EH) {
  int i = blockIdx.x * blockDim.x + threadIdx.x;
  if (i >= EH) return;
  const int e = i >> 2, h = i & 3;
  const int d = dst[e];
  float ex = expf(ebuf[i] - m[(size_t)d * HH + h]);
  ebuf[i] = ex;
  atomicAdd(&denom[(size_t)d * HH + h], ex);
}

// ---------------------------------------------------------------------------
// Message passing: out[dst] += (ex/denom[dst]) * hsrc[src].
// One thread per (edge, float4 chunk) -> coalesced hsrc gather and
// contention-free contiguous atomics within a wave.
// ---------------------------------------------------------------------------
__global__ __launch_bounds__(256) void egat_message(
    const int* __restrict__ src, const int* __restrict__ dst,
    const float* __restrict__ ebuf, const float* __restrict__ denom,
    const float4* __restrict__ hsrc, float* __restrict__ out, int EQ) {
  int i = blockIdx.x * blockDim.x + threadIdx.x;
  if (i >= EQ) return;
  const int e = i >> 6;           // 64 float4 chunks per edge (256 feats)
  const int q = i & 63;
  const int h = q >> 4;           // 16 chunks per head
  const int s = src[e], d = dst[e];
  const float a = ebuf[(size_t)e * HH + h] / denom[(size_t)d * HH + h];
  float4 v = hsrc[(size_t)s * 64 + q];
  float* o = out + ((size_t)d * 64 + q) * 4;
  atomicAdd(o + 0, a * v.x);
  atomicAdd(o + 1, a * v.y);
  atomicAdd(o + 2, a * v.z);
  atomicAdd(o + 3, a * v.w);
}

// ---------------------------------------------------------------------------
extern "C" void kernel_launch(void* const* d_in, const int* in_sizes, int n_in,
                              void* d_out, int out_size, void* d_ws, size_t ws_size,
                              hipStream_t stream) {
  const float* x      = (const float*)d_in[0];
  const float* efeats = (const float*)d_in[1];
  const int*   src    = (const int*)d_in[2];
  const int*   dst    = (const int*)d_in[3];
  const float* W_ni   = (const float*)d_in[4];
  const float* W_fij  = (const float*)d_in[5];
  const float* W_nj   = (const float*)d_in[6];
  const float* W_src  = (const float*)d_in[7];
  const float* attn   = (const float*)d_in[8];
  const float* bias   = (const float*)d_in[9];

  float* out  = (float*)d_out;
  float* fout = out + (size_t)NN * IN_F;        // second output: f [E, 64]

  // Workspace layout (floats): ni | nj | hsrc | ebuf | m | denom  (~36.5 MB)
  float* ws    = (float*)d_ws;
  float* ni    = ws;
  float* nj    = ni   + (size_t)NN * (HH * FEDG);
  float* hsrc  = nj   + (size_t)NN * (HH * FEDG);
  float* ebuf  = hsrc + (size_t)NN * (HH * FOUT);
  float* mbuf  = ebuf + (size_t)EEDG * HH;
  float* dbuf  = mbuf + (size_t)NN * HH;

  const int B = 256;

  // Phase 1: WMMA fp32 GEMMs (LDS-staged B, 64-column slices per block)
  {
    dim3 gN((NN / 16 + 7) / 8, 1);                 // 1250 strips -> 157 blocks
    egat_wmma_gemm<IN_F><<<gN, B, 0, stream>>>(x, W_ni, ni, NN, HH * FEDG);
    egat_wmma_gemm<IN_F><<<gN, B, 0, stream>>>(x, W_nj, nj, NN, HH * FEDG);
    dim3 gS((NN / 16 + 7) / 8, (HH * FOUT) / 64);  // 157 x 4
    egat_wmma_gemm<IN_F><<<gS, B, 0, stream>>>(x, W_src, hsrc, NN, HH * FOUT);
    dim3 gE((EEDG / 16) / 8, 1);                   // 20000 strips -> 2500 blocks
    // fij GEMM writes directly into the f output region (reused as scratch).
    egat_wmma_gemm<EIN_F><<<gE, B, 0, stream>>>(efeats, W_fij, fout, EEDG, HH * FEDG);
  }

  // Phase 2: residual init + edge softmax + aggregation
  const int nOut = NN * IN_F;                    // 5.12M
  const int nNH  = NN * HH;
  egat_init<<<(nOut + B - 1) / B, B, 0, stream>>>(x, out, mbuf, dbuf, nOut, nNH);

  egat_edge_scores<<<(EEDG + B - 1) / B, B, 0, stream>>>(
      src, dst, (const float4*)ni, (const float4*)nj,
      (const float4*)attn, (const float4*)bias,
      (float4*)fout, ebuf, EEDG);

  const int EH = EEDG * HH;
  egat_segmax<<<(EH + B - 1) / B, B, 0, stream>>>(dst, ebuf, mbuf, EH);
  egat_expsum<<<(EH + B - 1) / B, B, 0, stream>>>(dst, ebuf, mbuf, dbuf, EH);

  const int EQ = EEDG * 64;                      // edge x float4-chunks
  egat_message<<<(EQ + B - 1) / B, B, 0, stream>>>(
      src, dst, ebuf, dbuf, (const float4*)hsrc, out, EQ);
}